// MultiHeadCrossAttention_88897233092703
// MI455X (gfx1250) — compile-verified
//
#include <hip/hip_runtime.h>
#include <cstdint>

// ---------------------------------------------------------------------------
// MultiHeadCrossAttention decode step (q_len == 1) for MI455X / gfx1250.
//
// Algebraic collapse (q_len==1):
//   scores[b,h,s]  = key[b,s,:] . T[b,h,:] + c0[b,h] - cov*Wcov[h]
//     where T[b,h,:] = (1/SCALE) * Wk[:, h*64:(h+1)*64] @ qproj[b,h,:]
//   context[b,h,:] = (softmax-weighted sum of value rows) @ Wv[:,h-slice] + bv
// => key/value are each streamed from HBM exactly once (537 MB total,
//    ~23 us at 23.3 TB/s). Matrix work uses V_WMMA_F32_16X16X4_F32
//    (fp32-exact, wave32). Streaming uses double-buffered
//    global_load_async_to_lds_b128 (ASYNCcnt) to overlap HBM and WMMA.
// ---------------------------------------------------------------------------

#define B_   32
#define S_   4096
#define D_   512
#define H_   8
#define HD_  64
#define NS_  16            // S splits (flash-decoding style)
#define SC_  (S_ / NS_)    // 256 rows per split
#define STG_ 64            // rows staged in LDS at a time
#define NST_ (SC_ / STG_)  // 4 stages per split
#define KP_  516           // padded LDS row stride (floats): 516 % 64 == 4
#define SP_  260           // padded stride for score/E rows:  260 % 64 == 4
#define INVSCALE 0.125f    // 1/sqrt(64)

typedef __attribute__((ext_vector_type(2))) float v2f;
typedef __attribute__((ext_vector_type(4))) float v4f;
typedef __attribute__((ext_vector_type(8))) float v8f;
typedef __attribute__((ext_vector_type(4))) int   v4i;

// workspace layout (float offsets); total ~5.51M floats (~21 MB)
#define WS_T   0                           // B*16*512  scaled T (padded heads)
#define WS_C0  (WS_T  + B_*16*D_)          // B*16      per-head const (pad=-inf)
#define WS_L   (WS_C0 + B_*16)             // B*NS*8    per-split exp-sums
#define WS_E   (WS_L  + B_*NS_*H_)         // B*8*S     unnormalized exp scores
#define WS_U   (WS_E  + (size_t)B_*H_*S_)  // B*NS*16*512 partial weighted V sums

// gfx1250 async global->LDS copies (ASYNCcnt path); param types per probe:
// (v4i addrspace(1)*, v4i addrspace(3)*, imm offset, imm cpol)
#if defined(__gfx1250__) && __has_builtin(__builtin_amdgcn_global_load_async_to_lds_b128)
#define USE_ASYNC_LDS 1
typedef __attribute__((address_space(1))) v4i* gp4_t;
typedef __attribute__((address_space(3))) v4i* lp4_t;
#else
#define USE_ASYNC_LDS 0
#pragma message("gfx1250 async-to-LDS builtin not available; using manual staging")
#endif

// Issue one 64-row x 512-f32 stage (128 KB) into LDS (padded stride KP_).
__device__ __forceinline__ void stage_issue(const float* __restrict__ gsrc,
                                            float* lds, int t) {
  for (int idx = t; idx < STG_ * (D_ / 4); idx += 128) {
    const int r  = idx >> 7;          // row 0..63
    const int c4 = (idx & 127) * 4;   // col (float4 granularity)
#if USE_ASYNC_LDS
    __builtin_amdgcn_global_load_async_to_lds_b128(
        (gp4_t)(gsrc + (size_t)r * D_ + c4),
        (lp4_t)&lds[r * KP_ + c4], 0, 0);
#else
    v4f v = *(const v4f*)(gsrc + (size_t)r * D_ + c4);
    *(v4f*)&lds[r * KP_ + c4] = v;
#endif
  }
}

__device__ __forceinline__ void stage_wait() {
#if USE_ASYNC_LDS
#if __has_builtin(__builtin_amdgcn_s_wait_asynccnt)
  __builtin_amdgcn_s_wait_asynccnt(0);
#else
  asm volatile("s_wait_asynccnt 0" ::: "memory");
#endif
#endif
}

// ---------------------------------------------------------------------------
// Kernel 1: q projection + T matrix + c0 (tiny; ~1 MB of Wq/Wk traffic/block)
// ---------------------------------------------------------------------------
__global__ __launch_bounds__(256) void mha_prep(
    const float* __restrict__ query, const float* __restrict__ Wq,
    const float* __restrict__ bq,    const float* __restrict__ Wk,
    const float* __restrict__ bk,    float* __restrict__ ws) {
  const int b = blockIdx.x, t = threadIdx.x;
  __shared__ float qin[D_], qp[D_];
  for (int c = t; c < D_; c += 256) qin[c] = query[(size_t)b * D_ + c];
  __syncthreads();
  for (int c = t; c < D_; c += 256) {           // qproj = query @ Wq + bq
    float acc = bq[c];
    for (int d = 0; d < D_; ++d) acc += qin[d] * Wq[(size_t)d * D_ + c];
    qp[c] = acc;
  }
  __syncthreads();
  float* Tg = ws + WS_T + (size_t)b * 16 * D_;
  for (int idx = t; idx < 16 * D_; idx += 256) {
    const int h = idx >> 9, d = idx & (D_ - 1);
    float acc = 0.0f;
    if (h < H_) {
      const float* wkrow = Wk + (size_t)d * D_ + h * HD_;
      const float* qh = qp + h * HD_;
#pragma unroll 8
      for (int u = 0; u < HD_; ++u) acc += wkrow[u] * qh[u];
      acc *= INVSCALE;
    }
    Tg[idx] = acc;                               // padded heads: zero rows
  }
  if (t < 16) {
    float c0;
    if (t < H_) {
      float acc = 0.0f;
      for (int u = 0; u < HD_; ++u) acc += bk[t * HD_ + u] * qp[t * HD_ + u];
      c0 = acc * INVSCALE;
    } else {
      c0 = -INFINITY;                            // padded heads -> exp() == 0
    }
    ws[WS_C0 + b * 16 + t] = c0;
  }
}

// ---------------------------------------------------------------------------
// Kernel 2: per (batch, S-split): scores via WMMA, exp, partial sums l,
//           and partial attn-weighted value sums u — all fp32 WMMA,
//           double-buffered async HBM->LDS streaming.
// grid (NS_, B_), block 128 (= 4 wave32)
// ---------------------------------------------------------------------------
__global__ __launch_bounds__(128) void mha_main(
    const float* __restrict__ key, const float* __restrict__ value,
    const float* __restrict__ coverage, const unsigned char* __restrict__ mask,
    const float* __restrict__ wcov, float* __restrict__ ws) {
  const int j = blockIdx.x, b = blockIdx.y, t = threadIdx.x;
  const int lane = t & 31, wv = t >> 5;
  const int half = lane >> 4, l16 = lane & 15;
  const int s0 = j * SC_;

  __shared__ float Tlds[16 * KP_];      // A matrix (16 heads x 512), padded
  __shared__ float KV0[STG_ * KP_];     // double-buffered key/value stages
  __shared__ float KV1[STG_ * KP_];
  __shared__ float slds[16 * SP_];      // exp(score): 16 heads x 256 cols
  __shared__ float c0s[16], wcs[16];
  __shared__ float lpart[128];

  const float* keyb = key   + ((size_t)b * S_ + s0) * D_;
  const float* valb = value + ((size_t)b * S_ + s0) * D_;

  stage_issue(keyb, KV0, t);            // prefetch key stage 0 immediately

  // stage T (16x512) into LDS while key stage 0 is in flight
  const float* Tg = ws + WS_T + (size_t)b * 16 * D_;
  for (int idx = t; idx < 16 * (D_ / 4); idx += 128) {
    const int r = idx >> 7, c4 = (idx & 127) * 4;
    *(v4f*)&Tlds[r * KP_ + c4] = *(const v4f*)(Tg + (size_t)r * D_ + c4);
  }
  if (t < 16) c0s[t] = ws[WS_C0 + b * 16 + t];
  if (t < 16) wcs[t] = (t < H_) ? wcov[t] : 0.0f;

  // ---- phase K: scores = T (16x512) x key_chunk^T (512 x SC) ----
  for (int st = 0; st < NST_; ++st) {
    float* cur = (st & 1) ? KV1 : KV0;
    float* nxt = (st & 1) ? KV0 : KV1;
    stage_wait();                       // this wave's loads for stage st done
    __syncthreads();                    // all waves' loads + stage st-1 compute
    if (st + 1 < NST_)
      stage_issue(keyb + (size_t)(st + 1) * STG_ * D_, nxt, t);

    // wave wv owns the 16-row tile [r0 + wv*16, r0 + wv*16 + 16)
    const int r0 = st * STG_;
    v8f c = {0.f, 0.f, 0.f, 0.f, 0.f, 0.f, 0.f, 0.f};
    const int arow = l16 * KP_;                 // A: head row
    const int brow = (wv * 16 + l16) * KP_;     // B: key row (k = feature)
#pragma unroll 4
    for (int k0 = 0; k0 < D_; k0 += 4) {
      v2f a  = *(const v2f*)&Tlds[arow + k0 + 2 * half];
      v2f bf = *(const v2f*)&cur[brow + k0 + 2 * half];
      c = __builtin_amdgcn_wmma_f32_16x16x4_f32(false, a, false, bf,
                                                (short)0, c, false, false);
    }
    // epilogue: bias, coverage, mask, exp; E -> LDS; e -> workspace
    const int scol = r0 + wv * 16 + l16;        // column within this chunk
    const int sabs = s0 + scol;                 // absolute sequence position
    const float cov = coverage[(size_t)b * S_ + sabs];
    const int mk = mask[(size_t)b * S_ + sabs];
#pragma unroll
    for (int g = 0; g < 8; ++g) {
      const int hh = g + 8 * half;              // C layout: head = g + 8*half
      float sc = c[g] + c0s[hh] - cov * wcs[hh];
      sc = mk ? -INFINITY : sc;                 // branchless: exp(-inf) == 0
      const float e = __expf(sc);
      slds[hh * SP_ + scol] = e;
      if (half == 0)
        ws[WS_E + ((size_t)(b * H_ + g)) * S_ + sabs] = e;
    }
  }

  stage_issue(valb, KV0, t);            // prefetch value stage 0 during l-sums
  __syncthreads();

  // ---- per-head partial softmax denominators for this split ----
  {
    const int hh = t >> 4, i0 = t & 15;
    float p = 0.0f;
    if (hh < H_)
      for (int i = i0; i < SC_; i += 16) p += slds[hh * SP_ + i];
    lpart[t] = p;
  }
  __syncthreads();
  if (t < H_) {
    float l = 0.0f;
    for (int i = 0; i < 16; ++i) l += lpart[t * 16 + i];
    ws[WS_L + ((size_t)b * NS_ + j) * H_ + t] = l;
  }

  // ---- phase V: u (16 x 512) = E (16 x SC) x value_chunk (SC x 512) ----
  v8f acc[8];
#pragma unroll
  for (int i = 0; i < 8; ++i) acc[i] = (v8f){0.f,0.f,0.f,0.f,0.f,0.f,0.f,0.f};

  for (int st = 0; st < NST_; ++st) {
    float* cur = (st & 1) ? KV1 : KV0;
    float* nxt = (st & 1) ? KV0 : KV1;
    stage_wait();
    __syncthreads();
    if (st + 1 < NST_)
      stage_issue(valb + (size_t)(st + 1) * STG_ * D_, nxt, t);

    const int r0 = st * STG_;
#pragma unroll 2
    for (int kk = 0; kk < STG_; kk += 4) {
      v2f a = *(const v2f*)&slds[l16 * SP_ + r0 + kk + 2 * half];
#pragma unroll
      for (int i = 0; i < 8; ++i) {
        const int dbase = (wv * 8 + i) * 16;
        v2f bf;
        bf.x = cur[(kk + 2 * half) * KP_ + dbase + l16];
        bf.y = cur[(kk + 2 * half + 1) * KP_ + dbase + l16];
        acc[i] = __builtin_amdgcn_wmma_f32_16x16x4_f32(false, a, false, bf,
                                                       (short)0, acc[i],
                                                       false, false);
      }
    }
  }
  // write partial u (unnormalized); padded heads carry zeros
  float* up = ws + WS_U + ((size_t)b * NS_ + j) * 16 * D_;
#pragma unroll
  for (int i = 0; i < 8; ++i) {
    const int dbase = (wv * 8 + i) * 16;
#pragma unroll
    for (int g = 0; g < 8; ++g) {
      const int hh = g + 8 * half;
      up[(size_t)hh * D_ + dbase + l16] = acc[i][g];
    }
  }
}

// ---------------------------------------------------------------------------
// Kernel 3: normalize -> avg_attn, new_coverage. grid (S/256, B), block 256.
// ---------------------------------------------------------------------------
__global__ __launch_bounds__(256) void mha_attn(
    const float* __restrict__ coverage, float* __restrict__ out,
    const float* __restrict__ ws) {
  const int b = blockIdx.y, t = threadIdx.x;
  __shared__ float invL[H_];
  if (t < H_) {
    float l = 0.0f;
    for (int jj = 0; jj < NS_; ++jj)
      l += ws[WS_L + ((size_t)b * NS_ + jj) * H_ + t];
    invL[t] = 1.0f / l;
  }
  __syncthreads();
  const int s = blockIdx.x * 256 + t;
  float a = 0.0f;
#pragma unroll
  for (int h = 0; h < H_; ++h)
    a += ws[WS_E + ((size_t)(b * H_ + h)) * S_ + s] * invL[h];
  a *= (1.0f / H_);
  const float cov = coverage[(size_t)b * S_ + s];
  out[(size_t)B_ * D_ + (size_t)b * S_ + s] = a;                         // avg_attn
  out[(size_t)B_ * D_ + (size_t)B_ * S_ + (size_t)b * S_ + s] = cov + a; // new_cov
}

// ---------------------------------------------------------------------------
// Kernel 4: reduce u over splits, context = u@Wv + bv, output = ctx@Wo + bo.
// grid B, block 256.
// ---------------------------------------------------------------------------
__global__ __launch_bounds__(256) void mha_out(
    const float* __restrict__ Wv, const float* __restrict__ bv,
    const float* __restrict__ Wo, const float* __restrict__ bo,
    float* __restrict__ out, const float* __restrict__ ws) {
  const int b = blockIdx.x, t = threadIdx.x;
  __shared__ float invL[H_];
  __shared__ float usum[H_ * D_];
  __shared__ float ctx[D_];
  if (t < H_) {
    float l = 0.0f;
    for (int jj = 0; jj < NS_; ++jj)
      l += ws[WS_L + ((size_t)b * NS_ + jj) * H_ + t];
    invL[t] = 1.0f / l;
  }
  __syncthreads();
  for (int idx = t; idx < H_ * D_; idx += 256) {
    const int h = idx >> 9, d = idx & (D_ - 1);
    float a = 0.0f;
    for (int jj = 0; jj < NS_; ++jj)
      a += ws[WS_U + (((size_t)b * NS_ + jj) * 16 + h) * D_ + d];
    usum[idx] = a * invL[h];
  }
  __syncthreads();
  for (int c = t; c < D_; c += 256) {           // context = u @ Wv + bv
    const int h = c >> 6;
    float a = bv[c];
    const float* uh = usum + h * D_;
    for (int d = 0; d < D_; ++d) a += uh[d] * Wv[(size_t)d * D_ + c];
    ctx[c] = a;
  }
  __syncthreads();
  for (int o = t; o < D_; o += 256) {           // output = ctx @ Wo + bo
    float a = bo[o];
    for (int c = 0; c < D_; ++c) a += ctx[c] * Wo[(size_t)c * D_ + o];
    out[(size_t)b * D_ + o] = a;
  }
}

extern "C" void kernel_launch(void* const* d_in, const int* in_sizes, int n_in,
                              void* d_out, int out_size, void* d_ws, size_t ws_size,
                              hipStream_t stream) {
  const float* query    = (const float*)d_in[0];
  const float* key      = (const float*)d_in[1];
  const float* value    = (const float*)d_in[2];
  const float* coverage = (const float*)d_in[3];
  const unsigned char* mask = (const unsigned char*)d_in[4]; // numpy bool = 1 byte
  const float* Wq  = (const float*)d_in[5];
  const float* bq  = (const float*)d_in[6];
  const float* Wk  = (const float*)d_in[7];
  const float* bk  = (const float*)d_in[8];
  const float* Wv  = (const float*)d_in[9];
  const float* bv  = (const float*)d_in[10];
  const float* Wo  = (const float*)d_in[11];
  const float* bo  = (const float*)d_in[12];
  const float* Wcov = (const float*)d_in[13];
  float* out = (float*)d_out;
  float* ws  = (float*)d_ws;   // requires ~22 MB of scratch

  mha_prep<<<B_, 256, 0, stream>>>(query, Wq, bq, Wk, bk, ws);
  mha_main<<<dim3(NS_, B_), 128, 0, stream>>>(key, value, coverage, mask, Wcov, ws);
  mha_attn<<<dim3(S_ / 256, B_), 256, 0, stream>>>(coverage, out, ws);
  mha_out<<<B_, 256, 0, stream>>>(Wv, bv, Wo, bo, out, ws);
}